// GCN_27023934226807
// MI455X (gfx1250) — compile-verified
//
#include <hip/hip_runtime.h>

typedef float v2f __attribute__((ext_vector_type(2)));
typedef float v8f __attribute__((ext_vector_type(8)));

#define B_   64
#define N_   1024
#define E_   16384
#define D_   1024
#define C_   128
#define TOT  (B_ * N_)        // 65536
#define NCONV 4

// ---------- embed: h0[64,128] = relu(x[64,1024] @ W_emb[1024,128] + b_emb) ----------
__global__ __launch_bounds__(32) void embed_kernel(const float* __restrict__ x,
                                                   const float* __restrict__ W,
                                                   const float* __restrict__ bias,
                                                   float* __restrict__ h0) {
    const int lane = threadIdx.x;               // wave32
    const int m0 = blockIdx.x * 16;             // 4 M-tiles
    const int n0 = blockIdx.y * 16;             // 8 N-tiles
    const int mr = lane & 15;
    const int kh = (lane >> 4) << 1;            // 0 or 2
    const float* xr = x + (size_t)(m0 + mr) * D_;
    const float* wc = W + n0 + mr;
    v8f acc = {};
    for (int k0 = 0; k0 < D_; k0 += 4) {
        v2f a, b;
        a.x = xr[k0 + kh];
        a.y = xr[k0 + kh + 1];
        b.x = wc[(size_t)(k0 + kh) * C_];
        b.y = wc[(size_t)(k0 + kh + 1) * C_];
        acc = __builtin_amdgcn_wmma_f32_16x16x4_f32(false, a, false, b,
                                                    (short)0, acc, false, false);
    }
    const int nOut  = n0 + mr;
    const int mBase = m0 + (lane >> 4) * 8;     // D layout: lanes 0-15 -> M=r, 16-31 -> M=r+8
    const float bv = bias[nOut];
#pragma unroll
    for (int r = 0; r < 8; ++r) {
        float v = acc[r] + bv;
        h0[(size_t)(mBase + r) * C_ + nOut] = fmaxf(v, 0.0f);
    }
}

// ---------- broadcast: h[b*N+n, c] = h0[b, c] ----------
__global__ __launch_bounds__(256) void broadcast_kernel(const float* __restrict__ h0,
                                                        float* __restrict__ h) {
    const long i = (long)blockIdx.x * 256 + threadIdx.x;  // < TOT*C_
    const int c = (int)(i & (C_ - 1));
    const int b = (int)(i >> 17);                         // / (N_*C_) = / 2^17
    h[i] = h0[b * C_ + c];
}

// ---------- conv GEMM: hw[65536,128] = relu(h) @ W_conv[128,128] ----------
// 8 waves/block; each wave owns a 16-row M tile and ALL 8 N-tiles (16x128 output),
// so every A fragment is loaded once and feeds 8 independent back-to-back WMMAs.
// Full W_conv (64 KB) staged in LDS, shared by all waves of the block.
__global__ __launch_bounds__(256) void convgemm_kernel(const float* __restrict__ h,
                                                       const float* __restrict__ Wc,
                                                       float* __restrict__ hw) {
    __shared__ float Bt[C_ * C_];               // [k][n], 64 KB of the 320 KB WGP LDS
    const int tid = threadIdx.x;
    for (int i = tid; i < C_ * C_; i += 256)
        Bt[i] = Wc[i];
    __syncthreads();

    const int lane = tid & 31;
    const int wave = tid >> 5;
    const int m0 = blockIdx.x * 128 + wave * 16;
    const int mr = lane & 15;
    const int kh = (lane >> 4) << 1;
    const float* ar = h + (size_t)(m0 + mr) * C_;

    v8f acc[8] = {};
    for (int k0 = 0; k0 < C_; k0 += 4) {
        v2f a;
        a.x = fmaxf(ar[k0 + kh], 0.0f);         // ReLU folded into the A read
        a.y = fmaxf(ar[k0 + kh + 1], 0.0f);
#pragma unroll
        for (int nt = 0; nt < 8; ++nt) {
            v2f b;
            b.x = Bt[(k0 + kh) * C_ + nt * 16 + mr];
            b.y = Bt[(k0 + kh + 1) * C_ + nt * 16 + mr];
            acc[nt] = __builtin_amdgcn_wmma_f32_16x16x4_f32(false, a, false, b,
                                                            (short)0, acc[nt],
                                                            false, false);
        }
    }

    const int mBase = m0 + (lane >> 4) * 8;
#pragma unroll
    for (int nt = 0; nt < 8; ++nt) {
        const int nOut = nt * 16 + mr;
#pragma unroll
        for (int r = 0; r < 8; ++r)
            hw[(size_t)(mBase + r) * C_ + nOut] = acc[nt][r];
    }
}

// ---------- init h with the conv bias (scatter then accumulates on top) ----------
__global__ __launch_bounds__(256) void bias_init_kernel(float* __restrict__ p,
                                                        const float* __restrict__ bias) {
    const long i = (long)blockIdx.x * 256 + threadIdx.x;
    p[i] = bias[i & (C_ - 1)];
}

// ---------- scatter: h[dst] += hw[src] over all B*E batched edges ----------
// One wave per edge; each lane carries 4 columns (float4 gather, 4 f32 atomics in L2).
__global__ __launch_bounds__(256) void scatter_kernel(const long long* __restrict__ ei,
                                                      const float* __restrict__ hw,
                                                      float* __restrict__ h) {
    const long t = (long)blockIdx.x * 256 + threadIdx.x;
    const int  lane = (int)(t & 31);
    const long g = t >> 5;                       // batched edge id, < B*E
    const int  b = (int)(g >> 14);               // / E_ (2^14)
    const int  e = (int)(g & (E_ - 1));
    const int  src = (int)ei[e]      + b * N_;
    const int  dst = (int)ei[E_ + e] + b * N_;
    const float4 v = *(const float4*)(hw + (size_t)src * C_ + lane * 4);
    float* o = h + (size_t)dst * C_ + lane * 4;
    atomicAdd(o + 0, v.x);
    atomicAdd(o + 1, v.y);
    atomicAdd(o + 2, v.z);
    atomicAdd(o + 3, v.w);
}

// ---------- classifier: out[b] = relu(h[b*N + 0, :]) . W_cls + b_cls ----------
__global__ __launch_bounds__(32) void cls_kernel(const float* __restrict__ h,
                                                 const float* __restrict__ Wcls,
                                                 const float* __restrict__ bcls,
                                                 float* __restrict__ out) {
    const int b = blockIdx.x, lane = threadIdx.x;
    const float* row = h + (size_t)b * N_ * C_;   // node 0 of graph copy b
    float s = 0.0f;
#pragma unroll
    for (int j = 0; j < 4; ++j) {
        const int c = lane + 32 * j;
        s += fmaxf(row[c], 0.0f) * Wcls[c];       // final ReLU folded in
    }
#pragma unroll
    for (int off = 16; off > 0; off >>= 1) s += __shfl_xor(s, off, 32);
    if (lane == 0) out[b] = s + bcls[0];
}

extern "C" void kernel_launch(void* const* d_in, const int* in_sizes, int n_in,
                              void* d_out, int out_size, void* d_ws, size_t ws_size,
                              hipStream_t stream) {
    (void)in_sizes; (void)n_in; (void)out_size; (void)ws_size;
    const float*     x      = (const float*)d_in[0];
    const long long* ei     = (const long long*)d_in[1];   // int64 edge_index [2,E]
    const float*     W_emb  = (const float*)d_in[2];
    const float*     b_emb  = (const float*)d_in[3];
    const float*     W_conv = (const float*)d_in[4];
    const float*     b_conv = (const float*)d_in[5];
    const float*     W_cls  = (const float*)d_in[6];
    const float*     b_cls  = (const float*)d_in[7];
    float* out = (float*)d_out;

    char* ws = (char*)d_ws;
    float* h  = (float*)(ws);                                        // 32 MB
    float* hw = (float*)(ws + (size_t)TOT * C_ * sizeof(float));     // 32 MB
    float* h0 = (float*)(ws + (size_t)2 * TOT * C_ * sizeof(float)); // 32 KB

    const int elems_blocks = (TOT * C_) / 256;                  // 32768
    const int edge_blocks  = (int)(((long)B_ * E_ * 32) / 256); // 131072

    embed_kernel<<<dim3(4, 8), 32, 0, stream>>>(x, W_emb, b_emb, h0);
    broadcast_kernel<<<elems_blocks, 256, 0, stream>>>(h0, h);

    for (int i = 0; i < NCONV; ++i) {
        convgemm_kernel<<<TOT / 128, 256, 0, stream>>>(h, W_conv, hw);
        bias_init_kernel<<<elems_blocks, 256, 0, stream>>>(h, b_conv);
        scatter_kernel<<<edge_blocks, 256, 0, stream>>>(ei, hw, h);
    }

    cls_kernel<<<B_, 32, 0, stream>>>(h, W_cls, b_cls, out);
}